// AffineTransfer1_1_10986526343793
// MI455X (gfx1250) — compile-verified
//
#include <hip/hip_runtime.h>

#define C 256

typedef __attribute__((ext_vector_type(2))) float v2f;
typedef __attribute__((ext_vector_type(8))) float v8f;

// ---------------- utility kernels ----------------

__global__ void zero_kernel(float* __restrict__ p, long long n) {
  long long i = (long long)blockIdx.x * blockDim.x + threadIdx.x;
  long long stride = (long long)gridDim.x * blockDim.x;
  for (; i < n; i += stride) p[i] = 0.0f;
}

// Pack the weight column-blocks into contiguous [rows, 256] row-major matrices.
// Wcat1 [512,256] : rows 0-255 = W_lin_id,            rows 256-511 = W_tf_int[:,   0:256] (W0)
// Wcat2 [512,256] : rows 0-255 = W_tf_int[:,256:512], rows 256-511 = W_tf_inv[:,   0:256] (Wv1)
// Wcat3 [768,256] : rows 0-255 = W_tf_int[:,512:768], rows 256-511 = W_lin_inv,
//                   rows 512-767 = W_tf_inv[:,256:512] (Wv2)
__global__ void prepack_kernel(const float* __restrict__ Wint,
                               const float* __restrict__ Winv,
                               const float* __restrict__ Wlin_inv,
                               const float* __restrict__ Wlin_id,
                               float* __restrict__ Wcat1,
                               float* __restrict__ Wcat2,
                               float* __restrict__ Wcat3) {
  int r = blockIdx.x;
  int k = threadIdx.x;
  if (r < 512) {
    float v = (r < 256) ? Wlin_id[r * C + k] : Wint[(size_t)(r - 256) * 768 + k];
    Wcat1[(size_t)r * C + k] = v;
  } else if (r < 1024) {
    int rr = r - 512;
    float v = (rr < 256) ? Wint[(size_t)rr * 768 + 256 + k]
                         : Winv[(size_t)(rr - 256) * 512 + k];
    Wcat2[(size_t)rr * C + k] = v;
  } else {
    int rr = r - 1024;
    float v;
    if (rr < 256)      v = Wint[(size_t)rr * 768 + 512 + k];
    else if (rr < 512) v = Wlin_inv[(size_t)(rr - 256) * C + k];
    else               v = Winv[(size_t)(rr - 512) * 512 + 256 + k];
    Wcat3[(size_t)rr * C + k] = v;
  }
}

// x_inv accumulation: x_inv[d] += x[n], cnt[d] += 1   (64 threads, float4 lanes)
__global__ void dom_mean_acc_kernel(const float* __restrict__ x,
                                    const int* __restrict__ dom,
                                    float* __restrict__ xinv,
                                    float* __restrict__ cnt) {
  int n = blockIdx.x, c4 = threadIdx.x * 4;
  int d = dom[n];
  float4 v = *(const float4*)(x + (size_t)n * C + c4);
  float* dst = xinv + (size_t)d * C + c4;
  atomicAdd(dst + 0, v.x);
  atomicAdd(dst + 1, v.y);
  atomicAdd(dst + 2, v.z);
  atomicAdd(dst + 3, v.w);
  if (threadIdx.x == 0) atomicAdd(&cnt[d], 1.0f);
}

__global__ void dom_mean_div_kernel(float* __restrict__ xinv,
                                    const float* __restrict__ cnt) {
  int d = blockIdx.x, c4 = threadIdx.x * 4;
  float inv = 1.0f / fmaxf(cnt[d], 1.0f);
  float4* p = (float4*)(xinv + (size_t)d * C + c4);
  float4 v = *p;
  v.x *= inv; v.y *= inv; v.z *= inv; v.w *= inv;
  *p = v;
}

// msg[i] += x[src[e]]  for each edge e with intersect id i
__global__ void msg_acc_kernel(const float* __restrict__ x,
                               const int* __restrict__ nmap,  // [2,E]: src = [0:E]
                               const int* __restrict__ ii,
                               float* __restrict__ msg) {
  int e = blockIdx.x, c4 = threadIdx.x * 4;
  int s = nmap[e];
  int i = ii[e];
  float4 v = *(const float4*)(x + (size_t)s * C + c4);
  float* dst = msg + (size_t)i * C + c4;
  atomicAdd(dst + 0, v.x);
  atomicAdd(dst + 1, v.y);
  atomicAdd(dst + 2, v.z);
  atomicAdd(dst + 3, v.w);
}

// ---------------- WMMA GEMM ----------------
// out[row, col] = sum_k A[row,k] * W[col,k],  A:[16*rowTiles, 256], W:[16*colTiles, 256].
// Each wave computes a 16x64 output strip (4 accumulators sharing each A fetch):
// per K-step of 4 we issue 1 A load + 4 B loads + 4 x V_WMMA_F32_16X16X4_F32.
// Each 256-column block of the output goes to its own pointer (out0/out1/out2).
__global__ __launch_bounds__(128)
void wmma_gemm_kernel(const float* __restrict__ A, int rowTiles,
                      const float* __restrict__ W, int colQuads,
                      float* __restrict__ out0, float* __restrict__ out1,
                      float* __restrict__ out2) {
  int wave = threadIdx.x >> 5;
  int lane = threadIdx.x & 31;
  int strip = blockIdx.x * 4 + wave;
  if (strip >= rowTiles * colQuads) return;
  int rowTile = strip / colQuads;
  int q       = strip % colQuads;     // 64-column quad index
  int l16  = lane & 15;
  int half = lane >> 4;  // 0: K pair {0,1}, 1: K pair {2,3} within each step of 4

  // A 16x4 f32 layout: VGPR0 = K0 (lanes 0-15) / K2 (lanes 16-31); VGPR1 = K1 / K3.
  // B 4x16 f32 layout mirrors it with N striped across lanes.
  const float* aPtr = A + (size_t)(rowTile * 16 + l16) * C + half * 2;
  const float* wPtr = W + (size_t)(q * 64 + l16) * C + half * 2;  // + j*16*C per tile

  v8f acc0 = {}, acc1 = {}, acc2 = {}, acc3 = {};
#pragma unroll 8
  for (int k = 0; k < C; k += 4) {
    v2f a;  a.x  = aPtr[k];            a.y  = aPtr[k + 1];
    v2f b0; b0.x = wPtr[k];            b0.y = wPtr[k + 1];
    v2f b1; b1.x = wPtr[k + 16 * C];   b1.y = wPtr[k + 16 * C + 1];
    v2f b2; b2.x = wPtr[k + 32 * C];   b2.y = wPtr[k + 32 * C + 1];
    v2f b3; b3.x = wPtr[k + 48 * C];   b3.y = wPtr[k + 48 * C + 1];
    acc0 = __builtin_amdgcn_wmma_f32_16x16x4_f32(false, a, false, b0,
                                                 (short)0, acc0, false, false);
    acc1 = __builtin_amdgcn_wmma_f32_16x16x4_f32(false, a, false, b1,
                                                 (short)0, acc1, false, false);
    acc2 = __builtin_amdgcn_wmma_f32_16x16x4_f32(false, a, false, b2,
                                                 (short)0, acc2, false, false);
    acc3 = __builtin_amdgcn_wmma_f32_16x16x4_f32(false, a, false, b3,
                                                 (short)0, acc3, false, false);
  }

  // One 64-wide quad never crosses a 256-column block boundary.
  int colBlk  = (q * 64) >> 8;
  int colBase = (q * 64) & 255;
  float* outp = (colBlk == 0) ? out0 : ((colBlk == 1) ? out1 : out2);
  v8f accs[4] = {acc0, acc1, acc2, acc3};
#pragma unroll
  for (int j = 0; j < 4; ++j) {
    int colIn = colBase + j * 16 + l16;
#pragma unroll
    for (int r = 0; r < 8; ++r) {
      int mrow = rowTile * 16 + r + half * 8;  // C/D layout: VGPR r holds M=r / M=r+8
      outp[(size_t)mrow * C + colIn] = accs[j][r];
    }
  }
}

// ---------------- scatter kernels ----------------

// out[tgt[e]] += xw0[src[e]] + msgw1[ii[e]] + xinvw2[dmap0[ii[e]]]
__global__ void edge_scatter_kernel(const int* __restrict__ nmap, int E_,
                                    const int* __restrict__ ii,
                                    const int* __restrict__ dmap,
                                    const float* __restrict__ xw0,
                                    const float* __restrict__ msgw1,
                                    const float* __restrict__ xinvw2,
                                    float* __restrict__ out) {
  int e = blockIdx.x, c4 = threadIdx.x * 4;
  int s  = nmap[e];
  int t  = nmap[E_ + e];
  int i  = ii[e];
  int d0 = dmap[i];
  float4 a = *(const float4*)(xw0    + (size_t)s  * C + c4);
  float4 b = *(const float4*)(msgw1  + (size_t)i  * C + c4);
  float4 g = *(const float4*)(xinvw2 + (size_t)d0 * C + c4);
  float* dst = out + (size_t)t * C + c4;
  atomicAdd(dst + 0, a.x + b.x + g.x);
  atomicAdd(dst + 1, a.y + b.y + g.y);
  atomicAdd(dst + 2, a.z + b.z + g.z);
  atomicAdd(dst + 3, a.w + b.w + g.w);
}

// invacc[dmap1[m]] += msgv1[m] + xinvwv2[dmap0[m]]
__global__ void dom_scatter_kernel(const int* __restrict__ dmap, int M_,
                                   const float* __restrict__ msgv1,
                                   const float* __restrict__ xinvwv2,
                                   float* __restrict__ invacc) {
  int m = blockIdx.x, c4 = threadIdx.x * 4;
  int d0 = dmap[m];
  int d1 = dmap[M_ + m];
  float4 a = *(const float4*)(msgv1   + (size_t)m  * C + c4);
  float4 b = *(const float4*)(xinvwv2 + (size_t)d0 * C + c4);
  float* dst = invacc + (size_t)d1 * C + c4;
  atomicAdd(dst + 0, a.x + b.x);
  atomicAdd(dst + 1, a.y + b.y);
  atomicAdd(dst + 2, a.z + b.z);
  atomicAdd(dst + 3, a.w + b.w);
}

// out[n] += invacc[domain_indicator[n]]
__global__ void final_add_kernel(const int* __restrict__ dom,
                                 const float* __restrict__ invacc,
                                 float* __restrict__ out) {
  int n = blockIdx.x, c4 = threadIdx.x * 4;
  int d = dom[n];
  float4 a = *(const float4*)(invacc + (size_t)d * C + c4);
  float4* p = (float4*)(out + (size_t)n * C + c4);
  float4 v = *p;
  v.x += a.x; v.y += a.y; v.z += a.z; v.w += a.w;
  *p = v;
}

// ---------------- launcher ----------------

extern "C" void kernel_launch(void* const* d_in, const int* in_sizes, int n_in,
                              void* d_out, int out_size, void* d_ws, size_t ws_size,
                              hipStream_t stream) {
  const float* x        = (const float*)d_in[0];
  const int*   dom      = (const int*)d_in[1];
  const int*   nmap     = (const int*)d_in[2];   // [2,E]
  const int*   ii       = (const int*)d_in[3];   // [E]
  const int*   dmap     = (const int*)d_in[4];   // [2,M]
  const float* Wint     = (const float*)d_in[5]; // [256,768]
  const float* Winv     = (const float*)d_in[6]; // [256,512]
  const float* Wlin_inv = (const float*)d_in[7]; // [256,256]
  const float* Wlin_id  = (const float*)d_in[8]; // [256,256]

  const int N_ = in_sizes[0] / C;     // 131072
  const int E_ = in_sizes[3];         // 262144
  const int M_ = in_sizes[4] / 2;     // 65536
  const int D_ = 16384;

  float* out = (float*)d_out;

  // workspace layout (floats). msg, x_inv, cnt are contiguous for one zero pass.
  float* ws      = (float*)d_ws;
  float* msg     = ws;                               // M*256
  float* x_inv   = msg     + (size_t)M_ * C;         // D*256
  float* cnt     = x_inv   + (size_t)D_ * C;         // D
  float* xw0     = cnt     + (size_t)D_;             // N*256
  float* msgw1   = xw0     + (size_t)N_ * C;         // M*256
  float* msgv1   = msgw1   + (size_t)M_ * C;         // M*256
  float* xinvw2  = msgv1   + (size_t)M_ * C;         // D*256
  float* invacc  = xinvw2  + (size_t)D_ * C;         // D*256
  float* xinvwv2 = invacc  + (size_t)D_ * C;         // D*256
  float* Wcat1   = xinvwv2 + (size_t)D_ * C;         // 512*256
  float* Wcat2   = Wcat1   + (size_t)512 * C;        // 512*256
  float* Wcat3   = Wcat2   + (size_t)512 * C;        // 768*256

  // 0) zero the atomic accumulators (msg, x_inv, cnt are contiguous)
  long long nzero = (long long)M_ * C + (long long)D_ * C + D_;
  zero_kernel<<<4096, 256, 0, stream>>>(msg, nzero);

  // 1) pack weight blocks
  prepack_kernel<<<1792, 256, 0, stream>>>(Wint, Winv, Wlin_inv, Wlin_id,
                                           Wcat1, Wcat2, Wcat3);

  // 2) domain mean -> x_inv
  dom_mean_acc_kernel<<<N_, 64, 0, stream>>>(x, dom, x_inv, cnt);
  dom_mean_div_kernel<<<D_, 64, 0, stream>>>(x_inv, cnt);

  // 3) msg = segment_sum(x[src], intersect_indicator)
  msg_acc_kernel<<<E_, 64, 0, stream>>>(x, nmap, ii, msg);

  // 4) GEMMs (fp32 WMMA, 16x64 strip per wave).
  // GEMM1: x @ [Wid;W0]^T -> (out, xw0)
  {
    int rowTiles = N_ / 16, colQuads = 512 / 64;
    int strips = rowTiles * colQuads;
    wmma_gemm_kernel<<<(strips + 3) / 4, 128, 0, stream>>>(
        x, rowTiles, Wcat1, colQuads, out, xw0, xw0);
  }
  // GEMM2: msg @ [W1;Wv1]^T -> (msgw1, msgv1)
  {
    int rowTiles = M_ / 16, colQuads = 512 / 64;
    int strips = rowTiles * colQuads;
    wmma_gemm_kernel<<<(strips + 3) / 4, 128, 0, stream>>>(
        msg, rowTiles, Wcat2, colQuads, msgw1, msgv1, msgv1);
  }
  // GEMM3: x_inv @ [W2;Wlin_inv;Wv2]^T -> (xinvw2, invacc, xinvwv2)
  {
    int rowTiles = D_ / 16, colQuads = 768 / 64;
    int strips = rowTiles * colQuads;
    wmma_gemm_kernel<<<(strips + 3) / 4, 128, 0, stream>>>(
        x_inv, rowTiles, Wcat3, colQuads, xinvw2, invacc, xinvwv2);
  }

  // 5) edge scatter into out
  edge_scatter_kernel<<<E_, 64, 0, stream>>>(nmap, E_, ii, dmap,
                                             xw0, msgw1, xinvw2, out);

  // 6) message -> domain scatter into invacc
  dom_scatter_kernel<<<M_, 64, 0, stream>>>(dmap, M_, msgv1, xinvwv2, invacc);

  // 7) out[n] += invacc[dom[n]]
  final_add_kernel<<<N_, 64, 0, stream>>>(dom, invacc, out);
}